// MLP_model_67181878444575
// MI455X (gfx1250) — compile-verified
//
#include <hip/hip_runtime.h>
#include <hip/hip_bf16.h>
#include <math.h>

typedef __attribute__((ext_vector_type(2))) float v2f;
typedef __attribute__((ext_vector_type(8))) float v8f;

#define Bsz   256
#define Tlen  1200
#define Cch   116
#define HID   512
#define NPAIR 6670     // 116*115/2
#define KPAD  6672     // NPAIR rounded up to multiple of 4
#define TK    120      // timestep chunk held in LDS (multiple of 4; 1200/120 = 10 chunks)
#define XSTR  122      // LDS row stride in floats; 122 % 64 = 58 -> conflict-free frag loads
#define NCHUNK (Tlen / TK)

// LDS layout (floats). Covariance (stride 116, upper triangle only) overlays the
// x-staging region [0, 128*122) after a barrier; mu/std live past both regions.
#define COV_STR  Cch
#define XREGION  (128 * XSTR)                 // 15616
#define MU_OFF   XREGION                      // 15616
#define STD_OFF  (MU_OFF + Cch)               // 15732
#define SMEM_FLT (STD_OFF + Cch)              // 15848 floats = 63392 B  (< 64 KB)

// upper-triangle (incl. diagonal) 16x16 tile map for the 8x8 tile grid: 36 tiles
__device__ __constant__ unsigned char kTI[36] = {
    0,0,0,0,0,0,0,0, 1,1,1,1,1,1,1, 2,2,2,2,2,2, 3,3,3,3,3, 4,4,4,4, 5,5,5, 6,6, 7};
__device__ __constant__ unsigned char kTJ[36] = {
    0,1,2,3,4,5,6,7, 1,2,3,4,5,6,7, 2,3,4,5,6,7, 3,4,5,6,7, 4,5,6,7, 5,6,7, 6,7, 7};

// ---------------------------------------------------------------------------
// Kernel 1: per-batch correlation features via f32 WMMA SYRK.
// One block (256 thr = 8 wave32) per batch. Means are accumulated from the
// LDS-resident chunks, so the batch slice is read from HBM exactly once.
// ---------------------------------------------------------------------------
__global__ __launch_bounds__(256) void corr_feats_kernel(
    const float* __restrict__ in, float* __restrict__ feats)
{
  __shared__ __align__(16) float smem[SMEM_FLT];
  const int tid  = threadIdx.x;
  const int lane = tid & 31;
  const int w    = tid >> 5;
  const int lr   = lane & 15;
  const int half = lane >> 4;
  const int b    = blockIdx.x;
  const float* __restrict__ xb = in + (size_t)b * Tlen * Cch;

  // zero the channel-pad rows (116..127) of the x staging region once
  for (int e = tid; e < (128 - Cch) * TK; e += 256) {
    int c = Cch + e / TK, t = e - (c - Cch) * TK;
    smem[c * XSTR + t] = 0.f;
  }

  // hoist this wave's tile coordinates into registers (constant-indexed after unroll)
  int tmi[5], tni[5];
#pragma unroll
  for (int it = 0; it < 5; ++it) {
    const int tl = w + 8 * it;
    const int cl = (tl < 36) ? tl : 35;
    tmi[it] = kTI[cl];
    tni[it] = kTJ[cl];
  }

  v8f acc[5];
#pragma unroll
  for (int i = 0; i < 5; ++i)
#pragma unroll
    for (int r = 0; r < 8; ++r) acc[i][r] = 0.f;

  float musum = 0.f;   // per-channel running sum (threads < Cch)

  // --- SYRK on raw x: acc = sum_t x[c][t] * x[d][t], chunked through LDS ---
  for (int ch = 0; ch < NCHUNK; ++ch) {
    __syncthreads();                   // previous chunk fully consumed
    const float* __restrict__ xc = xb + (size_t)ch * TK * Cch;
    for (int e = tid; e < TK * Cch; e += 256) {
      int t = e / Cch, c = e - t * Cch;          // consecutive tid -> consecutive c
      smem[c * XSTR + t] = xc[t * Cch + c];      // transposed store
    }
    __syncthreads();

#pragma unroll
    for (int it = 0; it < 5; ++it) {
      if (w + 8 * it < 36) {           // wave-uniform guard: EXEC all-ones inside
        // f32 A frag (16x4): lane holds row lane%16, K pair 2*(lane/16)+{0,1}.
        // B = A^T of the other tile -> identical per-lane LDS pattern, ni base.
        const float* pa = &smem[(tmi[it] * 16 + lr) * XSTR + 2 * half];
        const float* pb = &smem[(tni[it] * 16 + lr) * XSTR + 2 * half];
        v8f c = acc[it];
#pragma unroll
        for (int kk = 0; kk < TK; kk += 4) {
          v2f a  = *(const v2f*)(pa + kk);
          v2f bf = *(const v2f*)(pb + kk);
          c = __builtin_amdgcn_wmma_f32_16x16x4_f32(false, a, false, bf,
                                                    (short)0, c, false, false);
        }
        acc[it] = c;
      }
    }

    // fold per-channel means out of the LDS-resident chunk (no extra HBM read)
    if (tid < Cch) {
      const float* xr = &smem[tid * XSTR];
      float s = 0.f;
#pragma unroll 8
      for (int t = 0; t < TK; ++t) s += xr[t];
      musum += s;
    }
  }

  if (tid < Cch) smem[MU_OFF + tid] = musum * (1.0f / (float)Tlen);

  // --- spill raw sums into compact cov region (overlays x region) ---
  __syncthreads();
#pragma unroll
  for (int it = 0; it < 5; ++it) {
    if (w + 8 * it < 36) {
      v8f c = acc[it];
      const int col = tni[it] * 16 + lr;
#pragma unroll
      for (int r = 0; r < 8; ++r) {
        const int row = tmi[it] * 16 + r + 8 * half;  // C/D layout: M = r + 8*(lane/16)
        if (row < Cch && col < Cch) smem[row * COV_STR + col] = c[r];
      }
    }
  }
  __syncthreads();

  // --- std from diagonal, with mean correction ---
  if (tid < Cch) {
    float mu  = smem[MU_OFF + tid];
    float raw = smem[tid * COV_STR + tid];
    float var = (raw - (float)Tlen * mu * mu) * (1.0f / (float)(Tlen - 1));
    smem[STD_OFF + tid] = sqrtf(fmaxf(var, 1e-30f));
  }
  __syncthreads();

  // --- clipped correlations, scattered in triu(k=1) packed order ---
  float* __restrict__ fout = feats + (size_t)b * KPAD;
  if (tid < Cch - 1) {
    const int i = tid;
    const float mui = smem[MU_OFF + i];
    const float si  = smem[STD_OFF + i];
    const int base  = i * (Cch - 1) - (i * (i - 1)) / 2;
    for (int j = i + 1; j < Cch; ++j) {
      float raw = smem[i * COV_STR + j];
      float num = raw - (float)Tlen * mui * smem[MU_OFF + j];
      float den = (float)(Tlen - 1) * si * smem[STD_OFF + j];
      float v = num / den;
      v = fminf(1.0f, fmaxf(-1.0f, v));
      fout[base + (j - i - 1)] = v;
    }
  }
  if (tid == Cch) { fout[NPAIR] = 0.f; fout[NPAIR + 1] = 0.f; }  // K padding
}

// ---------------------------------------------------------------------------
// Kernel 2: H = feats(256 x 6672) @ W1(6670 x 512) + b1, f32 WMMA, frags from
// global (working set fits in 192 MB L2). grid (16,8), 4 waves/block, one
// 16x16 tile per wave.
// ---------------------------------------------------------------------------
__global__ __launch_bounds__(128) void mlp1_kernel(
    const float* __restrict__ feats, const float* __restrict__ W1,
    const float* __restrict__ b1, float* __restrict__ H)
{
  const int lane = threadIdx.x & 31;
  const int w    = threadIdx.x >> 5;
  const int lr   = lane & 15;
  const int half = lane >> 4;
  const int m0   = blockIdx.x * 16;
  const int n0   = blockIdx.y * 64 + w * 16;

  v8f acc;
  const float bias = b1[n0 + lr];                 // acc element N = lane%16
#pragma unroll
  for (int r = 0; r < 8; ++r) acc[r] = bias;

  const float* __restrict__ pa = feats + (size_t)(m0 + lr) * KPAD + 2 * half;
  const float* __restrict__ pb = W1 + (size_t)(2 * half) * HID + n0 + lr;

  int k = 0;
  for (; k < NPAIR - 2; k += 4) {                 // k = 0 .. 6664 (rows <= 6667 valid)
    v2f a = *(const v2f*)(pa + k);
    v2f bb;
    bb.x = pb[(size_t)k * HID];
    bb.y = pb[(size_t)(k + 1) * HID];
    acc = __builtin_amdgcn_wmma_f32_16x16x4_f32(false, a, false, bb,
                                                (short)0, acc, false, false);
  }
  {                                               // tail k = 6668: guard W1 rows >= 6670;
    v2f a = *(const v2f*)(pa + k);                // A pad cols 6670/6671 are zero anyway
    v2f bb;
    bb.x = (half == 0) ? pb[(size_t)k * HID] : 0.f;
    bb.y = (half == 0) ? pb[(size_t)(k + 1) * HID] : 0.f;
    acc = __builtin_amdgcn_wmma_f32_16x16x4_f32(false, a, false, bb,
                                                (short)0, acc, false, false);
  }
#pragma unroll
  for (int r = 0; r < 8; ++r)
    H[(size_t)(m0 + r + 8 * half) * HID + n0 + lr] = acc[r];
}

// ---------------------------------------------------------------------------
// Kernel 3: logits = H(256 x 512) @ W2(512 x 2) + b2. Tiny; plain VALU.
// ---------------------------------------------------------------------------
__global__ __launch_bounds__(256) void mlp2_kernel(
    const float* __restrict__ H, const float* __restrict__ W2,
    const float* __restrict__ b2, float* __restrict__ out)
{
  const int b = blockIdx.x * blockDim.x + threadIdx.x;
  if (b >= Bsz) return;
  float a0 = b2[0], a1 = b2[1];
  const float* __restrict__ h = H + (size_t)b * HID;
  for (int n = 0; n < HID; ++n) {
    float hv = h[n];
    a0 = fmaf(hv, W2[n * 2 + 0], a0);
    a1 = fmaf(hv, W2[n * 2 + 1], a1);
  }
  out[b * 2 + 0] = a0;
  out[b * 2 + 1] = a1;
}

extern "C" void kernel_launch(void* const* d_in, const int* in_sizes, int n_in,
                              void* d_out, int out_size, void* d_ws, size_t ws_size,
                              hipStream_t stream) {
  const float* in = (const float*)d_in[0];   // (256,1200,116)
  const float* W1 = (const float*)d_in[1];   // (6670,512)
  const float* b1 = (const float*)d_in[2];   // (512,)
  const float* W2 = (const float*)d_in[3];   // (512,2)
  const float* b2 = (const float*)d_in[4];   // (2,)
  float* feats = (float*)d_ws;                          // 256*6672 floats
  float* H     = feats + (size_t)Bsz * KPAD;            // 256*512 floats
  float* out   = (float*)d_out;                         // 256*2 floats

  corr_feats_kernel<<<Bsz, 256, 0, stream>>>(in, feats);
  mlp1_kernel<<<dim3(Bsz / 16, HID / 64), 128, 0, stream>>>(feats, W1, b1, H);
  mlp2_kernel<<<1, 256, 0, stream>>>(H, W2, b2, out);
}